// VectorQuantizer_46523085750426
// MI455X (gfx1250) — compile-verified
//
#include <hip/hip_runtime.h>
#include <math.h>

typedef float v2f __attribute__((ext_vector_type(2)));
typedef float v8f __attribute__((ext_vector_type(8)));
typedef unsigned int v4u __attribute__((ext_vector_type(4)));
typedef int v4i __attribute__((ext_vector_type(4)));
typedef int v8i __attribute__((ext_vector_type(8)));

#define D        128
#define K        1024
#define TOK_WG   128
#define THREADS  256
#define WAVES    8
#define CHUNK    16
#define NCHUNK   (K / CHUNK)      // 64
#define TPAD     136              // x tile LDS row stride (floats): 2*136 % 64 == 16 -> conflict-free halves
#define BPAD     132              // codebook chunk LDS row stride (floats): 4*l16 banks, conflict-free b64
#define NTOK     (32 * 64 * 64)   // 131072 tokens
#define HW       4096
#define EPS_F    1e-5f

// d_out layout (floats), tuple concatenated in reference return order
#define OUT_Q    0
#define OUT_LOSS 16777216
#define OUT_IDX  16777217
#define OUT_NEWE 16908289

// d_ws layout (floats)
#define WS_ET    0          // eT[k][d]  (131072)
#define WS_ENORM 131072     // ||e_k||^2 (1024)
#define WS_SUMS  132096     // per-code sum of x vectors (131072)
#define WS_CNT   263168     // per-code counts (1024)

// ---- Tensor Data Mover: build D# (groups 0/1, zero-filled trailing groups) and issue load ----
// Group0: count=1, lds_addr, 57b global_addr, type=2. Group1: data_size=4B,
// pad_enable, pad_interval/amount, tensor dims, tile dims, dim0 stride.
__device__ __forceinline__ void tdm_load_2d(unsigned lds_off, const void* gptr,
                                            unsigned tensor_d0, unsigned tensor_d1,
                                            unsigned tile_d0, unsigned tile_d1,
                                            unsigned stride0,
                                            unsigned pad_int_code, unsigned pad_amt_code) {
  unsigned long long ga = (unsigned long long)(size_t)gptr;
  v4u g0;
  g0.x = 1u;                                                   // count=1, user mode
  g0.y = lds_off;                                              // lds_addr (bytes)
  g0.z = (unsigned)ga;                                         // global_addr[31:0]
  g0.w = (unsigned)((ga >> 32) & 0x01FFFFFFull) | (2u << 30);  // global_addr[56:32] | type=2
  v8i g1;
  g1[0] = (int)((2u << 16) |                                   // data_size = 4 bytes
                (1u << 20) |                                   // pad_enable
                (pad_int_code << 22) | (pad_amt_code << 25));
  g1[1] = (int)((tensor_d0 & 0xFFFFu) << 16);                  // tensor_dim0[15:0]
  g1[2] = (int)(((tensor_d0 >> 16) & 0xFFFFu) | ((tensor_d1 & 0xFFFFu) << 16));
  g1[3] = (int)(((tensor_d1 >> 16) & 0xFFFFu) | ((tile_d0 & 0xFFFFu) << 16));
  g1[4] = (int)(tile_d1 & 0xFFFFu);                            // tile_dim1, tile_dim2=0
  g1[5] = (int)stride0;                                        // tensor_dim0_stride[31:0]
  g1[6] = 0;                                                   // stride0 hi, stride1 lo
  g1[7] = 0;
  v4i z4 = {0, 0, 0, 0};
  v8i z8 = {0, 0, 0, 0, 0, 0, 0, 0};
  __builtin_amdgcn_tensor_load_to_lds(g0, g1, z4, z4, z8, 0);
}

// ---------------- k0: transpose codebook, ||e||^2, zero accumulators ----------------
__global__ __launch_bounds__(128) void vq_prep(const float* __restrict__ e,
                                               float* __restrict__ ws,
                                               float* __restrict__ out) {
  const int k = blockIdx.x;    // code
  const int d = threadIdx.x;   // dim
  float v = e[(size_t)d * K + k];
  ws[WS_ET + (size_t)k * D + d] = v;
  ws[WS_SUMS + (size_t)k * D + d] = 0.0f;
  float s = v * v;
#pragma unroll
  for (int off = 16; off >= 1; off >>= 1) s += __shfl_xor(s, off, 32);
  __shared__ float red[4];
  if ((threadIdx.x & 31) == 0) red[threadIdx.x >> 5] = s;
  __syncthreads();
  if (threadIdx.x == 0) {
    ws[WS_ENORM + k] = red[0] + red[1] + red[2] + red[3];
    ws[WS_CNT + k] = 0.0f;
    if (blockIdx.x == 0) out[OUT_LOSS] = 0.0f;
  }
}

// ---------------- k1: TDM-staged WMMA distance GEMM + argmin + gather + stats ----------------
__global__ __launch_bounds__(THREADS) void vq_main(const float* __restrict__ x,
                                                   float* __restrict__ ws,
                                                   float* __restrict__ out) {
  __shared__ float lds_x[D * TPAD];              // 69632 B : x tile [d][token] (TDM-padded)
  __shared__ float lds_b[2 * CHUNK * BPAD];      // 16896 B : double-buffered code chunk [code][d]
  __shared__ int   lds_idx[TOK_WG];
  __shared__ float lds_red[WAVES];

  const int tid  = threadIdx.x;
  const int wave = tid >> 5;
  const int lane = tid & 31;
  const int l16  = lane & 15;
  const int hi   = lane >> 4;        // wave32 half (per CDNA5 WMMA VGPR layouts)

  const int wgTok = blockIdx.x * TOK_WG;
  const int b     = wgTok >> 12;              // batch (constant per WG: 4096 % 128 == 0)
  const int hwB   = wgTok & (HW - 1);

  const float* eT    = ws + WS_ET;
  const float* enorm = ws + WS_ENORM;

  // ---- issue TDM: x tile (2D: 128 tokens x 128 dims, plane stride 4096), then chunk 0
  if (wave == 0) {
    tdm_load_2d((unsigned)(size_t)(void*)lds_x,
                x + (size_t)b * D * HW + hwB,
                /*tensor_d0=*/HW, /*tensor_d1=*/D,
                /*tile_d0=*/TOK_WG, /*tile_d1=*/D,
                /*stride0=*/HW,
                /*pad_int 128dw=*/6u, /*pad_amt 8dw=*/7u);
    tdm_load_2d((unsigned)(size_t)(void*)lds_b,
                eT,
                /*tensor_d0=*/NTOK, /*tensor_d1=*/1,
                /*tile_d0=*/CHUNK * D, /*tile_d1=*/1,
                /*stride0=*/NTOK,
                /*pad_int 128dw=*/6u, /*pad_amt 4dw=*/3u);
    __builtin_amdgcn_s_wait_tensorcnt(1);      // x tile complete (in-order per wave)
  }
  __syncthreads();

  // ---- A fragments for this wave's 16 tokens: A[M=l16][Kf = 4j + v + 2*hi]
  const int rowBase = wave * 16;
  const int col = rowBase + l16;
  v2f a[32];
#pragma unroll
  for (int j = 0; j < 32; ++j) {
    int k0 = 4 * j + 2 * hi;
    a[j].x = lds_x[k0 * TPAD + col];
    a[j].y = lds_x[(k0 + 1) * TPAD + col];
  }

  float best[8];
  int   bidx[8];
#pragma unroll
  for (int m = 0; m < 8; ++m) { best[m] = 3.0e38f; bidx[m] = 0; }

  for (int c = 0; c < NCHUNK; ++c) {
    if (wave == 0) __builtin_amdgcn_s_wait_tensorcnt(0);   // chunk c landed
    __syncthreads();                                       // visible to all; buf free for c+1
    if (wave == 0 && c + 1 < NCHUNK)                       // async prefetch next chunk
      tdm_load_2d((unsigned)(size_t)(void*)&lds_b[((c + 1) & 1) * CHUNK * BPAD],
                  eT + (size_t)(c + 1) * CHUNK * D,
                  NTOK, 1, CHUNK * D, 1, NTOK, 6u, 3u);

    v8f acc = {0.f, 0.f, 0.f, 0.f, 0.f, 0.f, 0.f, 0.f};
    const float* br = &lds_b[(c & 1) * CHUNK * BPAD + l16 * BPAD];  // B[Kf][N=l16]
#pragma unroll
    for (int j = 0; j < 32; ++j) {
      v2f bf = *(const v2f*)(br + 4 * j + 2 * hi);
      acc = __builtin_amdgcn_wmma_f32_16x16x4_f32(false, a[j], false, bf,
                                                  (short)0, acc, false, false);
    }
    const int   code = c * CHUNK + l16;
    const float en   = enorm[code];
#pragma unroll
    for (int m = 0; m < 8; ++m) {
      float score = en - 2.0f * acc[m];     // ||x||^2 constant per token: drop it
      if (score < best[m]) { best[m] = score; bidx[m] = code; }
    }
  }

  // min-reduce across the 16 lanes holding the same tokens (both wave halves)
#pragma unroll
  for (int off = 8; off >= 1; off >>= 1) {
#pragma unroll
    for (int m = 0; m < 8; ++m) {
      float od = __shfl_xor(best[m], off, 32);
      int   oi = __shfl_xor(bidx[m], off, 32);
      if (od < best[m] || (od == best[m] && oi < bidx[m])) { best[m] = od; bidx[m] = oi; }
    }
  }
#pragma unroll
  for (int m = 0; m < 8; ++m)
    if (l16 == m) lds_idx[rowBase + m + 8 * hi] = bidx[m];   // C/D row M = v + 8*hi
  __syncthreads();

  // epilogue: gather quantized, MSE partial, EMA stats
  float partial = 0.0f;
  float* sums = ws + WS_SUMS;
  float* cnts = ws + WS_CNT;
  for (int i = tid; i < TOK_WG * D; i += THREADS) {
    int t = i & (TOK_WG - 1);
    int d = i >> 7;
    int kb = lds_idx[t];
    float q  = eT[(size_t)kb * D + d];
    float xv = lds_x[d * TPAD + t];
    out[OUT_Q + (size_t)b * D * HW + (size_t)d * HW + hwB + t] = q;
    float df = xv - q;
    partial += df * df;
    atomicAdd(&sums[(size_t)kb * D + d], xv);
    if (d == 0) {
      atomicAdd(&cnts[kb], 1.0f);
      out[OUT_IDX + (size_t)(wgTok + t)] = (float)kb;
    }
  }
#pragma unroll
  for (int off = 16; off >= 1; off >>= 1) partial += __shfl_xor(partial, off, 32);
  if (lane == 0) lds_red[wave] = partial;
  __syncthreads();
  if (tid == 0) {
    float s = 0.0f;
    for (int w = 0; w < WAVES; ++w) s += lds_red[w];
    atomicAdd(&out[OUT_LOSS], s * (1.0f / 16777216.0f));
  }
}

// ---------------- k2: EMA codebook finalize (N_sum == NTOK is exact) ----------------
__global__ __launch_bounds__(256) void vq_newe(const float* __restrict__ ws,
                                               float* __restrict__ out) {
  int i = blockIdx.x * 256 + threadIdx.x;   // 0..131071
  int d = i >> 10;
  int k = i & (K - 1);
  float cnt = ws[WS_CNT + k];
  const float scale = (float)(131072.0 / (131072.0 + 1024.0 * 1e-5));  // N_sum/(N_sum+K*eps)
  float nst = (cnt + EPS_F) * scale;
  out[OUT_NEWE + (size_t)d * K + k] = ws[WS_SUMS + (size_t)k * D + d] / nst;
}

extern "C" void kernel_launch(void* const* d_in, const int* in_sizes, int n_in,
                              void* d_out, int out_size, void* d_ws, size_t ws_size,
                              hipStream_t stream) {
  const float* x = (const float*)d_in[0];
  const float* e = (const float*)d_in[1];
  float* out = (float*)d_out;
  float* ws  = (float*)d_ws;
  (void)in_sizes; (void)n_in; (void)out_size; (void)ws_size;

  vq_prep<<<K, 128, 0, stream>>>(e, ws, out);
  vq_main<<<NTOK / TOK_WG, THREADS, 0, stream>>>(x, ws, out);
  vq_newe<<<NTOK / 256, 256, 0, stream>>>(ws, out);
}